// ISSM_44375602102882
// MI455X (gfx1250) — compile-verified
//
#include <hip/hip_runtime.h>

// CDNA5 / gfx1250 wave32 Kalman-filter scan (ISSM), single-wave latency-optimized.
// D = A x B + C with V_WMMA_F32_16X16X4_F32; 16x16x16 matmul = 4 chained WMMAs.

typedef __attribute__((ext_vector_type(2))) float v2f;
typedef __attribute__((ext_vector_type(8))) float v8f;

#define HDIM 13
#define NP   16      // padded matrix dim
#define LDSP 17      // LDS row stride (bank-conflict-free)

// ---- WMMA fragment <-> LDS helpers (ISA 7.12.2 layouts) ----

// A-matrix 16x4 slice j (cols 4j..4j+3) of a 16x16 matrix.
// lane l: m = l&15, h = l>>4; elem v -> A[m][4j + 2h + v]
__device__ __forceinline__ v2f ldA(const float* M, int j, int m, int h) {
    v2f r;
    r.x = M[m * LDSP + (4 * j + 2 * h + 0)];
    r.y = M[m * LDSP + (4 * j + 2 * h + 1)];
    return r;
}
// B-matrix 4x16 slice j (rows 4j..4j+3); elem v -> B[4j + 2h + v][n]
__device__ __forceinline__ v2f ldB(const float* M, int j, int n, int h) {
    v2f r;
    r.x = M[(4 * j + 2 * h + 0) * LDSP + n];
    r.y = M[(4 * j + 2 * h + 1) * LDSP + n];
    return r;
}
// C/D 16x16: elem r -> D[r + 8h][n]
__device__ __forceinline__ void stD(float* M, v8f d, int n, int h) {
#pragma unroll
    for (int r = 0; r < 8; ++r) M[(r + 8 * h) * LDSP + n] = d[r];
}

// Full 16x16x16 matmul: D = A*B + C  (4 chained 16x16x4 WMMAs)
__device__ __forceinline__ v8f mm16(const float* A, const float* B, v8f c, int m, int n, int h) {
#pragma unroll
    for (int j = 0; j < 4; ++j) {
        v2f af = ldA(A, j, m, h);
        v2f bf = ldB(B, j, n, h);
        c = __builtin_amdgcn_wmma_f32_16x16x4_f32(false, af, false, bf,
                                                  (short)0, c, false, false);
    }
    return c;
}

__device__ __forceinline__ float wave_sum16(float q) {
    // lanes l and l+16 hold duplicate copies; xor-reduce within each 16-lane half
    q += __shfl_xor(q, 1, 32);
    q += __shfl_xor(q, 2, 32);
    q += __shfl_xor(q, 4, 32);
    q += __shfl_xor(q, 8, 32);
    return q;  // every lane now holds the 16-element sum
}

__global__ __launch_bounds__(32) void issm_kalman_scan(
    const float* __restrict__ z, const float* __restrict__ b,
    const float* __restrict__ F, const float* __restrict__ a,
    const float* __restrict__ g, const float* __restrict__ sigma,
    const float* __restrict__ m_prior, const float* __restrict__ S_prior,
    float* __restrict__ out, int T)
{
    __shared__ float sS [NP * LDSP];  // covariance (padded, zeros outside 13x13)
    __shared__ float sF [NP * LDSP];
    __shared__ float sFt[NP * LDSP];  // F^T
    __shared__ float sT1[NP * LDSP];  // temp product
    __shared__ float sM [NP * LDSP];  // I - K a^T
    __shared__ float sMt[NP * LDSP];  // (I - K a^T)^T
    __shared__ float sa[NP], sg[NP], sSa[NP], smu[NP], smuh[NP];

    const int lane = threadIdx.x;
    const int m16  = lane & 15;   // row (A) or col (B/C/D) index
    const int h    = lane >> 4;   // lane-half

    // zero-init (padding invariant: entries >= 13 stay zero forever)
    for (int i = lane; i < NP * LDSP; i += 32) {
        sS[i] = 0.f; sF[i] = 0.f; sFt[i] = 0.f; sT1[i] = 0.f; sM[i] = 0.f; sMt[i] = 0.f;
    }
    if (lane < NP) { sa[lane] = 0.f; sg[lane] = 0.f; sSa[lane] = 0.f; smu[lane] = 0.f; smuh[lane] = 0.f; }
    for (int i = lane; i < HDIM * HDIM; i += 32)
        sS[(i / HDIM) * LDSP + (i % HDIM)] = S_prior[i];
    if (lane < HDIM) smu[lane] = m_prior[lane];
    __syncthreads();

    float loss = 0.f;
    const float eps = 1e-8f;

    for (int t = 0; t < T; ++t) {
        // ---- stream in per-step params; prefetch ahead to hide latency ----
        const float* Fp = F + (size_t)t * (HDIM * HDIM);
        for (int i = lane; i < HDIM * HDIM; i += 32) {
            float v = Fp[i];
            int r0 = i / HDIM, c0 = i % HDIM;
            sF [r0 * LDSP + c0] = v;
            sFt[c0 * LDSP + r0] = v;
        }
        if (lane < HDIM) {
            sa[lane] = a[(size_t)t * HDIM + lane];
            sg[lane] = g[(size_t)t * HDIM + lane];
        }
        const float zt = z[t], bt = b[t], st = sigma[t];
        if (t + 8 < T) __builtin_prefetch(F + (size_t)(t + 8) * (HDIM * HDIM), 0, 1);
        __syncthreads();

        // ---- predict: S_hh = F S F^T + g g^T (skip at t==0) ----
        if (t > 0) {
            v8f zero = {};
            v8f t1 = mm16(sF, sS, zero, m16, m16, h);      // T1 = F * S
            stD(sT1, t1, m16, h);
            v8f cg;
            float gn = sg[m16];
#pragma unroll
            for (int r = 0; r < 8; ++r) cg[r] = sg[r + 8 * h] * gn;   // g g^T fragment
            v8f s2 = mm16(sT1, sFt, cg, m16, m16, h);      // S = T1 * F^T + g g^T
            stD(sS, s2, m16, h);
        }

        // ---- Sa = S_hh a  (row m16, split columns across lane halves) ----
        {
            float p = 0.f;
#pragma unroll
            for (int n0 = 0; n0 < 8; ++n0) {
                int n = 8 * h + n0;
                p += sS[m16 * LDSP + n] * sa[n];
            }
            p += __shfl_xor(p, 16, 32);
            if (h == 0) sSa[m16] = p;
        }
        // ---- mu_h = F mu_prev (or mu_prev at t==0) ----
        {
            float p;
            if (t > 0) {
                p = 0.f;
#pragma unroll
                for (int n0 = 0; n0 < 8; ++n0) {
                    int n = 8 * h + n0;
                    p += sF[m16 * LDSP + n] * smu[n];
                }
                p += __shfl_xor(p, 16, 32);
            } else {
                p = smu[m16];
            }
            if (h == 0) smuh[m16] = p;
        }

        // ---- innovation scalars ----
        float Svv  = wave_sum16(sa[m16] * sSa[m16]) + st * st;
        float aTmu = wave_sum16(sa[m16] * smuh[m16]);
        float delta = zt - bt - aTmu;
        float inv = 1.0f / (Svv + eps);
        loss += delta * delta;

        // ---- mu update: mu = mu_h + K * delta, K = Sa * inv ----
        if (h == 0) smu[m16] = smuh[m16] + (sSa[m16] * inv) * delta;

        // ---- build M = I - K a^T and M^T ----
        {
            float an = sa[m16];  // m16 = column n here
#pragma unroll
            for (int r = 0; r < 8; ++r) {
                int row = r + 8 * h;
                float kv = sSa[row] * inv;
                float v = ((row == m16) ? 1.f : 0.f) - kv * an;
                sM [row * LDSP + m16] = v;
                sMt[m16 * LDSP + row] = v;
            }
        }
        __syncthreads();

        // ---- Joseph form: S = M S_hh M^T + sigma^2 K K^T ----
        {
            v8f zero = {};
            v8f t1 = mm16(sM, sS, zero, m16, m16, h);      // T1 = M * S_hh
            stD(sT1, t1, m16, h);
            v8f ckk;
            float kn = sSa[m16] * inv;
            float s2 = st * st;
#pragma unroll
            for (int r = 0; r < 8; ++r) ckk[r] = s2 * (sSa[r + 8 * h] * inv) * kn;
            v8f snew = mm16(sT1, sMt, ckk, m16, m16, h);   // + sigma^2 K K^T
            stD(sS, snew, m16, h);
        }
        __syncthreads();
    }

    if (lane == 0) out[0] = loss;
}

extern "C" void kernel_launch(void* const* d_in, const int* in_sizes, int n_in,
                              void* d_out, int out_size, void* d_ws, size_t ws_size,
                              hipStream_t stream) {
    const float* z       = (const float*)d_in[0];
    const float* b       = (const float*)d_in[1];
    const float* F       = (const float*)d_in[2];
    const float* a       = (const float*)d_in[3];
    const float* g       = (const float*)d_in[4];
    const float* sigma   = (const float*)d_in[5];
    const float* m_prior = (const float*)d_in[6];
    const float* S_prior = (const float*)d_in[7];
    const int T = in_sizes[0];

    // Strictly sequential scan: one wave32 owns the whole recurrence.
    issm_kalman_scan<<<1, 32, 0, stream>>>(z, b, F, a, g, sigma,
                                           m_prior, S_prior, (float*)d_out, T);
}